// DyGrEncoder_75849122447503
// MI455X (gfx1250) — compile-verified
//
#include <hip/hip_runtime.h>
#include <hip/hip_bf16.h>

// ---------------------------------------------------------------------------
// DyGrEncoder for MI455X (gfx1250, wave32).
// All GEMMs run on V_WMMA_F32_16X16X4_F32 in NT orientation (contiguous
// 8-byte B-fragment loads -> global_load_b64). Each wave processes TWO
// 16-row tiles so every B fragment feeds two WMMAs.
// ---------------------------------------------------------------------------

#define NN 50000
#define EE 800000
#define DD 128
#define RB32 ((NN + 31) / 32)   // 1563 blocks of 32 rows (last block: 16 rows)

typedef __attribute__((ext_vector_type(2))) float v2f;
typedef __attribute__((ext_vector_type(8))) float v8f;

__device__ __forceinline__ float sigmoidf_(float x) {
    return 1.0f / (1.0f + expf(-x));
}

// acc0 += A0(16x128) * B^T tile ; acc1 += A1(16x128) * B^T tile (shared B frag)
// A tiles live in LDS (As = 32x128), B is global row-major [P][128],
// output columns are B rows [brow0 : brow0+16).
template <bool DUAL>
__device__ __forceinline__ void wmma_nt2(const float* __restrict__ As,
                                         const float* __restrict__ B,
                                         int brow0, v8f& acc0, v8f& acc1,
                                         int lane) {
    const int mn = lane & 15;          // A row / B row (=output col) for this lane
    const int kh = (lane >> 4) << 1;   // K sub-offset: 0 or 2
    const float* a0 = As + mn * 128 + kh;
    const float* a1 = As + (16 + mn) * 128 + kh;
    const float* bp = B + (size_t)(brow0 + mn) * 128 + kh;
#pragma unroll
    for (int k0 = 0; k0 < 128; k0 += 4) {
        v2f b;
        b.x = bp[k0];
        b.y = bp[k0 + 1];
        v2f a;
        a.x = a0[k0];
        a.y = a0[k0 + 1];
        acc0 = __builtin_amdgcn_wmma_f32_16x16x4_f32(false, a, false, b,
                                                     (short)0, acc0, false, false);
        if constexpr (DUAL) {
            v2f a2;
            a2.x = a1[k0];
            a2.y = a1[k0 + 1];
            acc1 = __builtin_amdgcn_wmma_f32_16x16x4_f32(false, a2, false, b,
                                                         (short)0, acc1, false, false);
        }
    }
}

// ---------------------------------------------------------------------------
// 128x128 transpose (conv weight -> NT orientation), trivial one-shot kernel
// ---------------------------------------------------------------------------
__global__ void __launch_bounds__(256) k_transpose128(const float* __restrict__ W,
                                                      float* __restrict__ Wt) {
    const int idx = blockIdx.x * 256 + threadIdx.x;   // 16384 elements
    const int r = idx >> 7;
    const int c = idx & 127;
    Wt[(size_t)c * 128 + r] = W[(size_t)r * 128 + c];
}

// ---------------------------------------------------------------------------
// m = x @ convW   as   x @ Wt^T   (Wt = convW transposed)
// ---------------------------------------------------------------------------
template <bool DUAL>
__device__ __forceinline__ void gemm_body(const float* As, const float* Wt,
                                          float* Mout, int row0, int wave,
                                          int lane) {
    const int col0 = wave * 16;
    v8f acc0 = {}, acc1 = {};
    wmma_nt2<DUAL>(As, Wt, col0, acc0, acc1, lane);
    const int hi = lane >> 4;
    const int n = lane & 15;
#pragma unroll
    for (int r = 0; r < 8; ++r) {
        const int m = r + (hi << 3);
        Mout[(size_t)(row0 + m) * 128 + col0 + n] = acc0[r];
        if constexpr (DUAL)
            Mout[(size_t)(row0 + 16 + m) * 128 + col0 + n] = acc1[r];
    }
}

__global__ void __launch_bounds__(256) k_gemm_nt(const float* __restrict__ Xin,
                                                 const float* __restrict__ Wt,
                                                 float* __restrict__ Mout) {
    __shared__ __align__(16) float As[32 * 128];
    const int row0 = blockIdx.x * 32;
    const bool dual = (row0 + 16) < NN;
    {
        const int n4 = dual ? 1024 : 512;
        const float4* s4 = (const float4*)(Xin + (size_t)row0 * 128);
        float4* d4 = (float4*)As;
        for (int i = threadIdx.x; i < n4; i += 256) d4[i] = s4[i];
    }
    __syncthreads();
    const int wave = threadIdx.x >> 5;
    const int lane = threadIdx.x & 31;
    if (dual)
        gemm_body<true>(As, Wt, Mout, row0, wave, lane);
    else
        gemm_body<false>(As, Wt, Mout, row0, wave, lane);
}

// ---------------------------------------------------------------------------
// agg = 0
// ---------------------------------------------------------------------------
__global__ void __launch_bounds__(256) k_zero4(float4* __restrict__ p, int n4) {
    int i = blockIdx.x * 256 + threadIdx.x;
    if (i < n4) p[i] = make_float4(0.f, 0.f, 0.f, 0.f);
}

// ---------------------------------------------------------------------------
// agg[dst] += m[src] * w   (one wave per edge, float4 per thread; agg is
// 25.6 MB -> L2-resident, atomics complete in L2)
// ---------------------------------------------------------------------------
__global__ void __launch_bounds__(256) k_scatter(const float* __restrict__ M,
                                                 const int* __restrict__ src,
                                                 const int* __restrict__ dst,
                                                 const float* __restrict__ ew,
                                                 float* __restrict__ agg) {
    long long tid = (long long)blockIdx.x * 256 + threadIdx.x;
    if (tid >= (long long)EE * 32) return;
    const int e = (int)(tid >> 5);
    const int c = ((int)tid & 31) << 2;
    const int s = src[e];
    const int d = dst[e];
    const float w = ew[e];
    const float4 mv = *(const float4*)(M + (size_t)s * 128 + c);
    float* ap = agg + (size_t)d * 128 + c;
    atomicAdd(ap + 0, mv.x * w);
    atomicAdd(ap + 1, mv.y * w);
    atomicAdd(ap + 2, mv.z * w);
    atomicAdd(ap + 3, mv.w * w);
}

// ---------------------------------------------------------------------------
// Fused GRU cell: x' = GRU(agg, x); r/z gates accumulate both GEMMs into
// shared WMMA accumulators, i_n/h_n kept separate (r gates h_n).
// ---------------------------------------------------------------------------
template <bool DUAL>
__device__ __forceinline__ void gru_body(const float* Ag, const float* Xs,
                                         const float* __restrict__ w_ih,
                                         const float* __restrict__ w_hh,
                                         const float* __restrict__ b_ih,
                                         const float* __restrict__ b_hh,
                                         float* __restrict__ xout, int row0,
                                         int wave, int lane) {
    const int col0 = wave * 16;
    v8f r0 = {}, r1 = {}, z0 = {}, z1 = {}, in0 = {}, in1 = {}, hn0 = {}, hn1 = {};
    wmma_nt2<DUAL>(Ag, w_ih, col0,       r0, r1, lane);
    wmma_nt2<DUAL>(Xs, w_hh, col0,       r0, r1, lane);
    wmma_nt2<DUAL>(Ag, w_ih, 128 + col0, z0, z1, lane);
    wmma_nt2<DUAL>(Xs, w_hh, 128 + col0, z0, z1, lane);
    wmma_nt2<DUAL>(Ag, w_ih, 256 + col0, in0, in1, lane);
    wmma_nt2<DUAL>(Xs, w_hh, 256 + col0, hn0, hn1, lane);

    const int n = lane & 15;
    const int hi = lane >> 4;
    const float br  = b_ih[col0 + n] + b_hh[col0 + n];
    const float bz  = b_ih[128 + col0 + n] + b_hh[128 + col0 + n];
    const float bin = b_ih[256 + col0 + n];
    const float bhn = b_hh[256 + col0 + n];
#pragma unroll
    for (int r = 0; r < 8; ++r) {
        const int m = r + (hi << 3);
        {
            const float rr = sigmoidf_(r0[r] + br);
            const float zz = sigmoidf_(z0[r] + bz);
            const float nn = tanhf(in0[r] + bin + rr * (hn0[r] + bhn));
            const float xv = Xs[m * 128 + col0 + n];
            xout[(size_t)(row0 + m) * 128 + col0 + n] = (1.f - zz) * nn + zz * xv;
        }
        if constexpr (DUAL) {
            const float rr = sigmoidf_(r1[r] + br);
            const float zz = sigmoidf_(z1[r] + bz);
            const float nn = tanhf(in1[r] + bin + rr * (hn1[r] + bhn));
            const float xv = Xs[(16 + m) * 128 + col0 + n];
            xout[(size_t)(row0 + 16 + m) * 128 + col0 + n] =
                (1.f - zz) * nn + zz * xv;
        }
    }
}

__global__ void __launch_bounds__(256) k_gru(const float* __restrict__ agg,
                                             const float* __restrict__ xin,
                                             const float* __restrict__ w_ih,
                                             const float* __restrict__ w_hh,
                                             const float* __restrict__ b_ih,
                                             const float* __restrict__ b_hh,
                                             float* __restrict__ xout) {
    __shared__ __align__(16) float Ag[32 * 128];
    __shared__ __align__(16) float Xs[32 * 128];
    const int row0 = blockIdx.x * 32;
    const bool dual = (row0 + 16) < NN;
    {
        const int n4 = dual ? 1024 : 512;
        const float4* a4 = (const float4*)(agg + (size_t)row0 * 128);
        const float4* x4 = (const float4*)(xin + (size_t)row0 * 128);
        float4* ad = (float4*)Ag;
        float4* xd = (float4*)Xs;
        for (int i = threadIdx.x; i < n4; i += 256) {
            ad[i] = a4[i];
            xd[i] = x4[i];
        }
    }
    __syncthreads();
    const int wave = threadIdx.x >> 5;
    const int lane = threadIdx.x & 31;
    if (dual)
        gru_body<true>(Ag, Xs, w_ih, w_hh, b_ih, b_hh, xout, row0, wave, lane);
    else
        gru_body<false>(Ag, Xs, w_ih, w_hh, b_ih, b_hh, xout, row0, wave, lane);
}

// ---------------------------------------------------------------------------
// Fused batched LSTM cell; all four gates accumulate both GEMMs into shared
// accumulators.
// ---------------------------------------------------------------------------
template <bool DUAL>
__device__ __forceinline__ void lstm_body(const float* Ts, const float* Hs,
                                          const float* __restrict__ c,
                                          const float* __restrict__ w_ih,
                                          const float* __restrict__ w_hh,
                                          const float* __restrict__ b_ih,
                                          const float* __restrict__ b_hh,
                                          float* __restrict__ hout,
                                          float* __restrict__ cout, int row0,
                                          int wave, int lane) {
    const int col0 = wave * 16;
    v8f i0 = {}, i1 = {}, f0 = {}, f1 = {}, g0 = {}, g1 = {}, o0 = {}, o1 = {};
    wmma_nt2<DUAL>(Ts, w_ih, col0,       i0, i1, lane);
    wmma_nt2<DUAL>(Hs, w_hh, col0,       i0, i1, lane);
    wmma_nt2<DUAL>(Ts, w_ih, 128 + col0, f0, f1, lane);
    wmma_nt2<DUAL>(Hs, w_hh, 128 + col0, f0, f1, lane);
    wmma_nt2<DUAL>(Ts, w_ih, 256 + col0, g0, g1, lane);
    wmma_nt2<DUAL>(Hs, w_hh, 256 + col0, g0, g1, lane);
    wmma_nt2<DUAL>(Ts, w_ih, 384 + col0, o0, o1, lane);
    wmma_nt2<DUAL>(Hs, w_hh, 384 + col0, o0, o1, lane);

    const int n = lane & 15;
    const int hi = lane >> 4;
    const float bi = b_ih[col0 + n] + b_hh[col0 + n];
    const float bf = b_ih[128 + col0 + n] + b_hh[128 + col0 + n];
    const float bg = b_ih[256 + col0 + n] + b_hh[256 + col0 + n];
    const float bo = b_ih[384 + col0 + n] + b_hh[384 + col0 + n];
#pragma unroll
    for (int r = 0; r < 8; ++r) {
        const int m = r + (hi << 3);
        {
            const size_t idx = (size_t)(row0 + m) * 128 + col0 + n;
            const float ig = sigmoidf_(i0[r] + bi);
            const float fg = sigmoidf_(f0[r] + bf);
            const float gg = tanhf(g0[r] + bg);
            const float og = sigmoidf_(o0[r] + bo);
            const float co = fg * c[idx] + ig * gg;
            cout[idx] = co;
            hout[idx] = og * tanhf(co);
        }
        if constexpr (DUAL) {
            const size_t idx = (size_t)(row0 + 16 + m) * 128 + col0 + n;
            const float ig = sigmoidf_(i1[r] + bi);
            const float fg = sigmoidf_(f1[r] + bf);
            const float gg = tanhf(g1[r] + bg);
            const float og = sigmoidf_(o1[r] + bo);
            const float co = fg * c[idx] + ig * gg;
            cout[idx] = co;
            hout[idx] = og * tanhf(co);
        }
    }
}

__global__ void __launch_bounds__(256) k_lstm(const float* __restrict__ ht,
                                              const float* __restrict__ h,
                                              const float* __restrict__ c,
                                              const float* __restrict__ w_ih,
                                              const float* __restrict__ w_hh,
                                              const float* __restrict__ b_ih,
                                              const float* __restrict__ b_hh,
                                              float* __restrict__ hout,
                                              float* __restrict__ cout) {
    __shared__ __align__(16) float Ts[32 * 128];
    __shared__ __align__(16) float Hs[32 * 128];
    const int row0 = blockIdx.x * 32;
    const bool dual = (row0 + 16) < NN;
    {
        const int n4 = dual ? 1024 : 512;
        const float4* t4 = (const float4*)(ht + (size_t)row0 * 128);
        const float4* h4 = (const float4*)(h + (size_t)row0 * 128);
        float4* td = (float4*)Ts;
        float4* hd = (float4*)Hs;
        for (int i = threadIdx.x; i < n4; i += 256) {
            td[i] = t4[i];
            hd[i] = h4[i];
        }
    }
    __syncthreads();
    const int wave = threadIdx.x >> 5;
    const int lane = threadIdx.x & 31;
    if (dual)
        lstm_body<true>(Ts, Hs, c, w_ih, w_hh, b_ih, b_hh, hout, cout, row0,
                        wave, lane);
    else
        lstm_body<false>(Ts, Hs, c, w_ih, w_hh, b_ih, b_hh, hout, cout, row0,
                         wave, lane);
}

// ---------------------------------------------------------------------------
extern "C" void kernel_launch(void* const* d_in, const int* in_sizes, int n_in,
                              void* d_out, int out_size, void* d_ws, size_t ws_size,
                              hipStream_t stream) {
    (void)in_sizes; (void)n_in; (void)out_size; (void)ws_size;

    const float* X     = (const float*)d_in[0];
    const int*   ei    = (const int*)d_in[1];
    const float* ew    = (const float*)d_in[2];
    const float* Hin   = (const float*)d_in[3];
    const float* Cin   = (const float*)d_in[4];
    const float* convW = (const float*)d_in[5];
    const float* gwih  = (const float*)d_in[6];
    const float* gwhh  = (const float*)d_in[7];
    const float* gbih  = (const float*)d_in[8];
    const float* gbhh  = (const float*)d_in[9];
    const float* lwih  = (const float*)d_in[10];
    const float* lwhh  = (const float*)d_in[11];
    const float* lbih  = (const float*)d_in[12];
    const float* lbhh  = (const float*)d_in[13];

    float* out    = (float*)d_out;
    float* Htilde = out;                           // [N,128]
    float* Hout   = out + (size_t)NN * DD;         // [N,128]
    float* Cout   = out + 2 * (size_t)NN * DD;     // [N,128]

    float* ws   = (float*)d_ws;
    float* Mbuf = ws;                              // [N,128]
    float* Agg  = ws + (size_t)NN * DD;            // [N,128]
    float* Xbuf = ws + 2 * (size_t)NN * DD;        // [N,128]
    float* Wt   = ws + 3 * (size_t)NN * DD;        // [128,128] transposed conv W

    const int* src = ei;        // edge_index[0]
    const int* dst = ei + EE;   // edge_index[1]

    const dim3 blk(256);
    const int zb = (NN * DD / 4 + 255) / 256;
    const int sb = (int)(((long long)EE * 32 + 255) / 256);

    // ----- GatedGraphConv layer 0 -----
    k_transpose128<<<64, blk, 0, stream>>>(convW, Wt);
    k_gemm_nt<<<RB32, blk, 0, stream>>>(X, Wt, Mbuf);
    k_zero4<<<zb, blk, 0, stream>>>((float4*)Agg, NN * DD / 4);
    k_scatter<<<sb, blk, 0, stream>>>(Mbuf, src, dst, ew, Agg);
    k_gru<<<RB32, blk, 0, stream>>>(Agg, X, gwih, gwhh, gbih, gbhh, Xbuf);

    // ----- GatedGraphConv layer 1 -----
    k_transpose128<<<64, blk, 0, stream>>>(convW + DD * DD, Wt);
    k_gemm_nt<<<RB32, blk, 0, stream>>>(Xbuf, Wt, Mbuf);
    k_zero4<<<zb, blk, 0, stream>>>((float4*)Agg, NN * DD / 4);
    k_scatter<<<sb, blk, 0, stream>>>(Mbuf, src, dst, ew, Agg);
    k_gru<<<RB32, blk, 0, stream>>>(Agg, Xbuf, gwih, gwhh, gbih, gbhh, Htilde);

    // ----- batched LSTM cell -----
    k_lstm<<<RB32, blk, 0, stream>>>(Htilde, Hin, Cin, lwih, lwhh, lbih, lbhh,
                                     Hout, Cout);
}